// SageGNN_87256555585790
// MI455X (gfx1250) — compile-verified
//
#include <hip/hip_runtime.h>
#include <hip/hip_bf16.h>

typedef float v2f __attribute__((ext_vector_type(2)));
typedef float v8f __attribute__((ext_vector_type(8)));

// Exact-fp32 WMMA: D(16x16,f32) = A(16x4,f32) * B(4x16,f32) + C
// args: (neg_a, A, neg_b, B, c_mod, C, reuse_a, reuse_b)
#define WMMA_F32X4(a, b, c) \
    __builtin_amdgcn_wmma_f32_16x16x4_f32(false, (a), false, (b), (short)0, (c), false, false)

// ---------------------------------------------------------------------------
// Degree kernels
// ---------------------------------------------------------------------------
__global__ __launch_bounds__(256) void deg_count_kernel(
    const long long* __restrict__ ei, float* __restrict__ deg, long long nedges) {
  long long e = (long long)blockIdx.x * blockDim.x + threadIdx.x;
  if (e < nedges) {
    long long dst = ei[nedges + e];
    atomicAdd(&deg[dst], 1.0f);
  }
}

__global__ __launch_bounds__(256) void deg_inv_kernel(float* __restrict__ deg, int n) {
  int i = blockIdx.x * blockDim.x + threadIdx.x;
  if (i < n) {
    float d = deg[i];
    deg[i] = 1.0f / (d < 1.0f ? 1.0f : d);
  }
}

// ---------------------------------------------------------------------------
// Edge scatter: agg[dst] += h[src]. 16 lanes per edge, float4 per lane (64 f).
// All tensors fit in the 192MB L2, so these atomics stay on-chip.
// ---------------------------------------------------------------------------
__global__ __launch_bounds__(256) void scatter_kernel(
    const long long* __restrict__ ei, const float* __restrict__ h_in,
    long long in_stride, float* __restrict__ agg, long long nedges) {
  long long idx = (long long)blockIdx.x * blockDim.x + threadIdx.x;
  long long e = idx >> 4;
  int l = (int)(idx & 15);
  if (e >= nedges) return;
  long long src = ei[e];
  long long dst = ei[nedges + e];
  const float4 v = *(const float4*)(h_in + src * in_stride + (long long)l * 4);
  float* ap = agg + dst * 64 + (long long)l * 4;
  atomicAdd(ap + 0, v.x);
  atomicAdd(ap + 1, v.y);
  atomicAdd(ap + 2, v.z);
  atomicAdd(ap + 3, v.w);
}

// ---------------------------------------------------------------------------
// SAGE layer: h_out = relu((agg .* inv_deg) @ Wl + bl + h_in @ Wr)
// One wave per 16-node row block; 4 column tiles of 16 -> 4 accumulators.
// Weights staged in LDS *K-pair interleaved*: element (k,n) lives at
//   sW[(k>>1)*128 + n*2 + (k&1)]
// so each lane's B fragment {W[k][n], W[k+1][n]} is one aligned ds_load_b64.
// ---------------------------------------------------------------------------
__global__ __launch_bounds__(256) void sage_layer_wmma_kernel(
    const float* __restrict__ h_in, long long in_stride,
    const float* __restrict__ agg, const float* __restrict__ inv_deg,
    const float* __restrict__ Wl, const float* __restrict__ bl,
    const float* __restrict__ Wr,
    float* __restrict__ h_out, long long out_stride, int nnodes) {
  __shared__ float sWl[64 * 64];
  __shared__ float sWr[64 * 64];

  const int tid = threadIdx.x;
  // Cooperative swizzled fill: pairIdx -> (p = k/2, n); read rows 2p and 2p+1
  // at column n (coalesced across threads), write one contiguous float2.
#pragma unroll
  for (int i = 0; i < (64 * 64 / 2) / 256; ++i) {
    const int pairIdx = tid + i * 256;
    const int p = pairIdx >> 6;      // k pair 0..31
    const int n = pairIdx & 63;      // column
    float2 vl, vr;
    vl.x = Wl[(2 * p) * 64 + n];
    vl.y = Wl[(2 * p + 1) * 64 + n];
    vr.x = Wr[(2 * p) * 64 + n];
    vr.y = Wr[(2 * p + 1) * 64 + n];
    *(float2*)&sWl[p * 128 + n * 2] = vl;
    *(float2*)&sWr[p * 128 + n * 2] = vr;
  }
  __syncthreads();

  const int wave = tid >> 5;
  const int lane = tid & 31;
  const int row16 = lane & 15;   // A: M row (both halves); B/D: N column
  const int khalf = lane >> 4;   // K offset 0 / 2 within a 4-wide step
  const long long nodeBase = ((long long)blockIdx.x * 8 + wave) * 16;
  if (nodeBase >= nnodes) return;  // wave-uniform: EXEC stays all-ones

  const float idg = inv_deg[nodeBase + row16];
  const float* aggRow = agg + (nodeBase + row16) * 64;
  const float* hRow = h_in + (nodeBase + row16) * in_stride;

  v8f c[4];
#pragma unroll
  for (int t = 0; t < 4; ++t) c[t] = (v8f){};

#pragma unroll
  for (int k0 = 0; k0 < 64; k0 += 4) {
    const int ka = k0 + khalf * 2;       // even
    const int pbase = (ka >> 1) * 128;   // swizzled pair row
    v2f aL, aR;
    aL.x = aggRow[ka] * idg;             // fold mean-normalization into A
    aL.y = aggRow[ka + 1] * idg;
    aR.x = hRow[ka];
    aR.y = hRow[ka + 1];
#pragma unroll
    for (int t = 0; t < 4; ++t) {
      const int n = t * 16 + row16;
      const v2f bL = *(const v2f*)&sWl[pbase + n * 2];  // one ds_load_b64
      const v2f bR = *(const v2f*)&sWr[pbase + n * 2];
      c[t] = WMMA_F32X4(aL, bL, c[t]);
      c[t] = WMMA_F32X4(aR, bR, c[t]);
    }
  }

  // Epilogue: bias + ReLU + store. D layout: VGPR r -> M=r (lanes 0-15),
  // M=r+8 (lanes 16-31); N = lane&15.
#pragma unroll
  for (int t = 0; t < 4; ++t) {
    const int col = t * 16 + row16;
    const float bv = bl[col];
#pragma unroll
    for (int r = 0; r < 8; ++r) {
      const long long node = nodeBase + r + (khalf ? 8 : 0);
      float v = c[t][r] + bv;
      v = v > 0.0f ? v : 0.0f;
      h_out[node * out_stride + col] = v;
    }
  }
}

// ---------------------------------------------------------------------------
// Final FC: out[N,32] = jk[N,192] @ fcW[192,32] + fcb
// Same K-pair-interleaved LDS layout: (k,n) -> sW[(k>>1)*64 + n*2 + (k&1)].
// ---------------------------------------------------------------------------
__global__ __launch_bounds__(256) void fc_wmma_kernel(
    const float* __restrict__ jk, const float* __restrict__ fcW,
    const float* __restrict__ fcb, float* __restrict__ out, int nnodes) {
  __shared__ float sW[192 * 32];
  const int tid = threadIdx.x;
#pragma unroll
  for (int i = 0; i < (192 * 32 / 2) / 256; ++i) {
    const int pairIdx = tid + i * 256;
    const int p = pairIdx >> 5;      // k pair 0..95
    const int n = pairIdx & 31;      // column
    float2 v;
    v.x = fcW[(2 * p) * 32 + n];
    v.y = fcW[(2 * p + 1) * 32 + n];
    *(float2*)&sW[p * 64 + n * 2] = v;
  }
  __syncthreads();

  const int wave = tid >> 5;
  const int lane = tid & 31;
  const int row16 = lane & 15;
  const int khalf = lane >> 4;
  const long long nodeBase = ((long long)blockIdx.x * 8 + wave) * 16;
  if (nodeBase >= nnodes) return;

  const float* jkRow = jk + (nodeBase + row16) * 192;

  v8f c[2];
  c[0] = (v8f){};
  c[1] = (v8f){};

#pragma unroll 8
  for (int k0 = 0; k0 < 192; k0 += 4) {
    const int ka = k0 + khalf * 2;
    const int pbase = (ka >> 1) * 64;
    v2f a;
    a.x = jkRow[ka];
    a.y = jkRow[ka + 1];
#pragma unroll
    for (int t = 0; t < 2; ++t) {
      const int n = t * 16 + row16;
      const v2f b = *(const v2f*)&sW[pbase + n * 2];
      c[t] = WMMA_F32X4(a, b, c[t]);
    }
  }

#pragma unroll
  for (int t = 0; t < 2; ++t) {
    const int col = t * 16 + row16;
    const float bv = fcb[col];
#pragma unroll
    for (int r = 0; r < 8; ++r) {
      const long long node = nodeBase + r + (khalf ? 8 : 0);
      out[node * 32 + col] = c[t][r] + bv;
    }
  }
}

// ---------------------------------------------------------------------------
// Launcher
// ---------------------------------------------------------------------------
extern "C" void kernel_launch(void* const* d_in, const int* in_sizes, int n_in,
                              void* d_out, int out_size, void* d_ws, size_t ws_size,
                              hipStream_t stream) {
  const int F = 64, H = 64;
  const int N = in_sizes[0] / F;         // 100000
  const long long E = in_sizes[1] / 2;   // 1200000

  const float* x = (const float*)d_in[0];
  const long long* ei = (const long long*)d_in[1];
  const float* Wl[3] = {(const float*)d_in[2], (const float*)d_in[5], (const float*)d_in[8]};
  const float* bl[3] = {(const float*)d_in[3], (const float*)d_in[6], (const float*)d_in[9]};
  const float* Wr[3] = {(const float*)d_in[4], (const float*)d_in[7], (const float*)d_in[10]};
  const float* fcW = (const float*)d_in[11];
  const float* fcb = (const float*)d_in[12];
  float* out = (float*)d_out;

  // Workspace: deg/inv_deg [N] | agg [N,64] | jk [N,192]  (~103 MB)
  float* ws = (float*)d_ws;
  float* deg = ws;
  float* agg = ws + N;
  float* jk = ws + N + (size_t)N * 64;

  // In-degree -> reciprocal (clipped to >= 1)
  hipMemsetAsync(deg, 0, (size_t)N * sizeof(float), stream);
  deg_count_kernel<<<(int)((E + 255) / 256), 256, 0, stream>>>(ei, deg, E);
  deg_inv_kernel<<<(N + 255) / 256, 256, 0, stream>>>(deg, N);

  const int layerBlocks = (N / 16 + 7) / 8;       // one wave per 16-node tile
  const int scatterBlocks = (int)((E * 16 + 255) / 256);

  const float* h = x;
  long long hstride = F;
  for (int i = 0; i < 3; ++i) {
    hipMemsetAsync(agg, 0, (size_t)N * 64 * sizeof(float), stream);
    scatter_kernel<<<scatterBlocks, 256, 0, stream>>>(ei, h, hstride, agg, E);
    float* hout = jk + (size_t)i * H;  // column block i of jk[N,192] (JK-cat for free)
    sage_layer_wmma_kernel<<<layerBlocks, 256, 0, stream>>>(
        h, hstride, agg, deg, Wl[i], bl[i], Wr[i], hout, 192, N);
    h = hout;
    hstride = 192;
  }

  fc_wmma_kernel<<<layerBlocks, 256, 0, stream>>>(jk, fcW, fcb, out, N);
}